// NeRFLoss_91164975824972
// MI455X (gfx1250) — compile-verified
//
#include <hip/hip_runtime.h>
#include <math.h>

// ---------------- problem constants (match reference) ----------------
#define NPIX   65536
#define NRAYS  65536
#define NCLS   256
#define DIM    16
#define L_OPAC 0.001f
#define L_DIST 0.001f
#define L_SAM  0.0001f

typedef __attribute__((ext_vector_type(2))) float v2f;
typedef __attribute__((ext_vector_type(8))) float v8f;

// ---------------- workspace layout (float offsets) ----------------
#define WS_COUNTS  0        // [256]   per-class pixel counts
#define WS_SUMS    256      // [4096]  per-class semantic sums
#define WS_U       4352     // [4096]  class means u
#define WS_DEVSUM  8448     // [256]   sum of ||s - u[cl]||
#define WS_VALIDCL 8704     // [256]   counts > MIN_PIXNUM (0/1)
#define WS_UPHI    8960     // [4096]  u / phi   (phi folded into B matrix)
#define WS_SCAL    13056    // [0]=proto_nce [1]=norm_num [2]=w_sum

// ==================================================================
// 0) zero the accumulated parts of the workspace (every launch)
// ==================================================================
__global__ void k_zero(float* ws) {
    int i = blockIdx.x * 256 + threadIdx.x;          // 4352 threads
    if (i < 4352) ws[WS_COUNTS + i] = 0.f;           // counts + sums
    if (i < 256)  ws[WS_DEVSUM + i] = 0.f;
    if (i < 4)    ws[WS_SCAL   + i] = 0.f;
}

// ==================================================================
// 1) rgb = (pred-gt)^2 ; opac = 0.001 * (-o*log(o))
// ==================================================================
__global__ void k_rgb_opac(const float* __restrict__ pred,
                           const float* __restrict__ gt,
                           const float* __restrict__ opac,
                           float* __restrict__ out_rgb,
                           float* __restrict__ out_opac) {
    int i = blockIdx.x * 256 + threadIdx.x;
    if (i < NPIX * 3) { float d = pred[i] - gt[i]; out_rgb[i] = d * d; }
    if (i < NPIX)     { float o = opac[i] + 1e-10f; out_opac[i] = L_OPAC * (-o * logf(o)); }
}

// ==================================================================
// 2) distortion loss: one thread per ray, serial prefix scan of 48
//    contiguous samples (each cacheline read exactly once).
// ==================================================================
__global__ void k_dist(const float* __restrict__ w_in,
                       const float* __restrict__ deltas,
                       const float* __restrict__ ts,
                       const int*   __restrict__ rays_a,
                       float* __restrict__ out_dist) {
    int r = blockIdx.x * 256 + threadIdx.x;
    if (r >= NRAYS) return;
    int start = rays_a[r * 3 + 1];
    int cnt   = rays_a[r * 3 + 2];
    float W = 0.f, WT = 0.f, acc = 0.f;
    for (int j = 0; j < cnt; ++j) {
        float w = w_in[start + j];
        float t = ts[start + j];
        float d = deltas[start + j];
        acc += 2.f * w * fmaf(t, W, -WT) + w * w * d * (1.f / 3.f);
        W  += w;
        WT  = fmaf(w, t, WT);
    }
    out_dist[r] = L_DIST * acc;
}

// ==================================================================
// 3) per-class counts + semantic sums (LDS-hierarchical atomics)
// ==================================================================
__global__ void k_seg_sums(const int* __restrict__ sam,
                           const float* __restrict__ sem,
                           float* __restrict__ ws) {
    __shared__ float lcnt[NCLS];
    __shared__ float lsum[NCLS * DIM];
    int tid = threadIdx.x;
    for (int i = tid; i < NCLS; i += 256)       lcnt[i] = 0.f;
    for (int i = tid; i < NCLS * DIM; i += 256) lsum[i] = 0.f;
    __syncthreads();
    for (int p = blockIdx.x * 256 + tid; p < NPIX; p += gridDim.x * 256) {
        int seg = sam[p];
        if (seg > 0) {
            int c = seg - 1;
            atomicAdd(&lcnt[c], 1.f);
            const float4* srow = (const float4*)(sem + p * DIM);
#pragma unroll
            for (int q = 0; q < 4; ++q) {
                float4 v = srow[q];
                atomicAdd(&lsum[c * DIM + 4 * q + 0], v.x);
                atomicAdd(&lsum[c * DIM + 4 * q + 1], v.y);
                atomicAdd(&lsum[c * DIM + 4 * q + 2], v.z);
                atomicAdd(&lsum[c * DIM + 4 * q + 3], v.w);
            }
        }
    }
    __syncthreads();
    for (int i = tid; i < NCLS; i += 256)       atomicAdd(&ws[WS_COUNTS + i], lcnt[i]);
    for (int i = tid; i < NCLS * DIM; i += 256) atomicAdd(&ws[WS_SUMS   + i], lsum[i]);
}

// ==================================================================
// 4) class means u = sums / max(counts,1)
// ==================================================================
__global__ void k_u(float* __restrict__ ws) {
    int i = blockIdx.x * 256 + threadIdx.x;          // 4096 threads
    if (i >= NCLS * DIM) return;
    float cnt = ws[WS_COUNTS + (i >> 4)];
    ws[WS_U + i] = ws[WS_SUMS + i] / fmaxf(cnt, 1.f);
}

// ==================================================================
// 5) dev = ||s - u[cl]|| segment-sum ; norm-loss partials
// ==================================================================
__global__ void k_dev(const int* __restrict__ sam,
                      const float* __restrict__ sem,
                      float* __restrict__ ws) {
    __shared__ float lu[NCLS * DIM];
    __shared__ float ldev[NCLS];
    __shared__ float lsc[2];
    int tid = threadIdx.x;
    for (int i = tid; i < NCLS * DIM; i += 256) lu[i] = ws[WS_U + i];
    for (int i = tid; i < NCLS; i += 256)       ldev[i] = 0.f;
    if (tid < 2) lsc[tid] = 0.f;
    __syncthreads();
    float nn = 0.f, wsm = 0.f;
    for (int p = blockIdx.x * 256 + tid; p < NPIX; p += gridDim.x * 256) {
        int seg = sam[p];
        if (seg > 0) {
            int c = seg - 1;
            float ss = 0.f, dd = 0.f;
#pragma unroll
            for (int d = 0; d < DIM; ++d) {
                float s = sem[p * DIM + d];
                ss = fmaf(s, s, ss);
                float df = s - lu[c * DIM + d];
                dd = fmaf(df, df, dd);
            }
            float nrm = sqrtf(ss);
            nn += (nrm - 1.f) * (nrm - 1.f);
            wsm += 1.f;
            atomicAdd(&ldev[c], sqrtf(dd));
        }
    }
    atomicAdd(&lsc[0], nn);
    atomicAdd(&lsc[1], wsm);
    __syncthreads();
    for (int i = tid; i < NCLS; i += 256) atomicAdd(&ws[WS_DEVSUM + i], ldev[i]);
    if (tid == 0) atomicAdd(&ws[WS_SCAL + 1], lsc[0]);
    if (tid == 1) atomicAdd(&ws[WS_SCAL + 2], lsc[1]);
}

// ==================================================================
// 6) phi, valid_cl, u_phi = u / phi  (phi folded into the B matrix)
// ==================================================================
__global__ void k_phi(float* __restrict__ ws) {
    int c = threadIdx.x;                              // one block, 256 threads
    float cnt   = ws[WS_COUNTS + c];
    float csafe = fmaxf(cnt, 1.f);
    float phi_raw = ws[WS_DEVSUM + c] / (csafe * logf(cnt + 10.f));
    float phi = fminf(fmaxf(phi_raw * 10.f, 0.1f), 1.f);
    float inv = 1.f / phi;
    ws[WS_VALIDCL + c] = (cnt > 2.f) ? 1.f : 0.f;     // MIN_PIXNUM == 2
#pragma unroll
    for (int d = 0; d < DIM; ++d)
        ws[WS_UPHI + c * DIM + d] = ws[WS_U + c * DIM + d] * inv;
}

// ==================================================================
// 7) WMMA NCE kernel: logits = semantic @ (u/phi)^T via chained
//    V_WMMA_F32_16X16X4_F32, fused with exp / row-sum / num pick /
//    -log(num/denom) reduction.  One wave = 16 rows x 256 classes.
//    __expf -> single hardware v_exp_f32 per element (vs ~12-op expf).
// ==================================================================
#define UST 17   // padded LDS stride for u rows -> conflict-free B loads

__global__ __launch_bounds__(256) void k_nce(const int* __restrict__ sam,
                                             const float* __restrict__ sem,
                                             const float* __restrict__ ws,
                                             float* __restrict__ proto) {
    __shared__ float lu[NCLS * UST];
    __shared__ float lv[NCLS];
    int tid = threadIdx.x;
    for (int i = tid; i < NCLS * DIM; i += 256)
        lu[(i >> 4) * UST + (i & 15)] = ws[WS_UPHI + i];
    for (int i = tid; i < NCLS; i += 256) lv[i] = ws[WS_VALIDCL + i];
    __syncthreads();

    int wv   = tid >> 5;
    int lane = tid & 31;
    int half = lane >> 4;        // 0: K even pair, 1: K odd pair (ISA A layout)
    int lh   = lane & 15;
    int m    = blockIdx.x * 128 + wv * 16;   // 8 waves * 16 rows = 128 rows/blk

    // ---- A fragments: row m+lh, K-chunk kk holds K = 4kk+2*half .. +1
    const float* srow = sem + (m + lh) * DIM;
    v2f a[4];
#pragma unroll
    for (int kk = 0; kk < 4; ++kk) {
        int k0 = 4 * kk + 2 * half;
        a[kk] = (v2f){srow[k0], srow[k0 + 1]};
    }

    // class ids of my 8 accumulator rows (D layout: VGPR r <-> row half*8+r)
    int clr[8];
#pragma unroll
    for (int r = 0; r < 8; ++r)
        clr[r] = sam[m + half * 8 + r] - 1;          // negative if invalid

    float den[8], num[8];
#pragma unroll
    for (int r = 0; r < 8; ++r) { den[r] = 0.f; num[r] = 0.f; }

#pragma unroll 2
    for (int t = 0; t < 16; ++t) {                   // 16 class tiles
        int c = 16 * t + lh;
        const float* urow = &lu[c * UST];
        v2f b[4];
#pragma unroll
        for (int kk = 0; kk < 4; ++kk) {
            int k0 = 4 * kk + 2 * half;
            b[kk] = (v2f){urow[k0], urow[k0 + 1]};
        }
        v8f acc = {0.f, 0.f, 0.f, 0.f, 0.f, 0.f, 0.f, 0.f};
        acc = __builtin_amdgcn_wmma_f32_16x16x4_f32(false, a[0], false, b[0], (short)0, acc, false, false);
        acc = __builtin_amdgcn_wmma_f32_16x16x4_f32(false, a[1], false, b[1], (short)0, acc, false, false);
        acc = __builtin_amdgcn_wmma_f32_16x16x4_f32(false, a[2], false, b[2], (short)0, acc, false, false);
        acc = __builtin_amdgcn_wmma_f32_16x16x4_f32(false, a[3], false, b[3], (short)0, acc, false, false);
        float vcl = lv[c];
#pragma unroll
        for (int r = 0; r < 8; ++r) {
            float e = __expf(acc[r]) * vcl;          // hw v_exp_f32 path
            den[r] += e;
            if (clr[r] == c) num[r] = e;             // exactly one hit per row
        }
    }

    // half-wave (16-lane) sum reduction; masks <16 keep lanes in their half
#pragma unroll
    for (int r = 0; r < 8; ++r) {
#pragma unroll
        for (int s = 1; s < 16; s <<= 1) {
            den[r] += __shfl_xor(den[r], s);
            num[r] += __shfl_xor(num[r], s);
        }
    }

    if (lh == 0) {                                   // lanes 0 and 16
        float lsum = 0.f;
#pragma unroll
        for (int r = 0; r < 8; ++r) {
            int i  = m + half * 8 + r;
            int sg = sam[i];
            if (sg > 0 && lv[sg - 1] > 0.f) {
                float dn = den[r] + 1e-6f;
                lsum += -logf(num[r] / dn);
            }
        }
        atomicAdd(proto, lsum);
    }
}

// ==================================================================
// 8) finalize scalar semantic loss
// ==================================================================
__global__ void k_final(const float* __restrict__ ws, float* __restrict__ out_sem) {
    if (threadIdx.x == 0 && blockIdx.x == 0) {
        float nce = ws[WS_SCAL + 0];
        float nn  = ws[WS_SCAL + 1];
        float wsm = ws[WS_SCAL + 2];
        out_sem[0] = L_SAM * nce + 100.f * (nn / fmaxf(wsm, 1.f));
    }
}

// ==================================================================
extern "C" void kernel_launch(void* const* d_in, const int* in_sizes, int n_in,
                              void* d_out, int out_size, void* d_ws, size_t ws_size,
                              hipStream_t stream) {
    const float* rgb_pred = (const float*)d_in[0];
    const float* rgb_gt   = (const float*)d_in[1];
    const float* opacity  = (const float*)d_in[2];
    const float* ws_in    = (const float*)d_in[3];
    const float* deltas   = (const float*)d_in[4];
    const float* ts       = (const float*)d_in[5];
    const int*   rays_a   = (const int*)d_in[6];
    const int*   sam      = (const int*)d_in[7];
    const float* semantic = (const float*)d_in[8];

    float* out      = (float*)d_out;
    float* out_rgb  = out;                 // 196608
    float* out_opac = out + 196608;        // 65536
    float* out_dist = out + 262144;        // 65536
    float* out_sem  = out + 327680;        // 1

    float* w = (float*)d_ws;

    k_zero    <<<17, 256, 0, stream>>>(w);
    k_rgb_opac<<<(NPIX * 3 + 255) / 256, 256, 0, stream>>>(rgb_pred, rgb_gt, opacity,
                                                           out_rgb, out_opac);
    k_dist    <<<(NRAYS + 255) / 256, 256, 0, stream>>>(ws_in, deltas, ts, rays_a, out_dist);
    k_seg_sums<<<64, 256, 0, stream>>>(sam, semantic, w);
    k_u       <<<16, 256, 0, stream>>>(w);
    k_dev     <<<64, 256, 0, stream>>>(sam, semantic, w);
    k_phi     <<<1, 256, 0, stream>>>(w);
    k_nce     <<<NPIX / 128, 256, 0, stream>>>(sam, semantic, w, w + WS_SCAL);
    k_final   <<<1, 1, 0, stream>>>(w, out_sem);
}